// BatchedGAT_10557029614167
// MI455X (gfx1250) — compile-verified
//
#include <hip/hip_runtime.h>
#include <hip/hip_bf16.h>

typedef __attribute__((ext_vector_type(2))) float v2f;
typedef __attribute__((ext_vector_type(4))) float v4f;
typedef __attribute__((ext_vector_type(8))) float v8f;

#define GAT_B   4
#define GAT_N   4096
#define GAT_K   32
#define GAT_IN  64
#define GAT_H   4
#define GAT_D   64
#define GAT_HD  256
#define GAT_NN  (GAT_B * GAT_N)   // 16384 rows
#define NEG_SLOPE 0.2f
#define LN_EPS    1e-5f

// ---------------------------------------------------------------------------
// Kernel 1: xl = x@Wl + bl, xr = x@Wr + br via V_WMMA_F32_16X16X4_F32.
// One wave computes one 16x16 output tile of BOTH xl and xr (A-frag reuse).
// A-matrix 16x4 fp32 layout (ISA 7.12.2): lane m in 0..15 holds K=k0,k0+1 in
// v0,v1; lanes 16..31 hold K=k0+2,k0+3. B-matrix 4x16: rows striped across
// lanes within a VGPR (mirrored grouping). C/D: c[v] = (M = v + 8*(lane/16),
// N = lane%16).
// ---------------------------------------------------------------------------
__global__ __launch_bounds__(32) void gat_proj_wmma(
    const float* __restrict__ x,
    const float* __restrict__ Wl, const float* __restrict__ bl,
    const float* __restrict__ Wr, const float* __restrict__ br,
    float* __restrict__ xl, float* __restrict__ xr)
{
    const int m0   = blockIdx.x * 16;     // row tile   (0..16383 step 16)
    const int n0   = blockIdx.y * 16;     // col tile   (0..255 step 16)
    const int lane = threadIdx.x;
    const int mrow = lane & 15;           // M (for A) / N (for B,C,D)
    const int grp  = lane >> 4;           // K-pair select

    v8f cl = {};
    v8f cr = {};

    #pragma unroll
    for (int k0 = 0; k0 < GAT_IN; k0 += 4) {
        const int ka = k0 + 2 * grp;
        v2f a;
        a.x = x[(size_t)(m0 + mrow) * GAT_IN + ka];
        a.y = x[(size_t)(m0 + mrow) * GAT_IN + ka + 1];
        v2f bL, bR;
        bL.x = Wl[(size_t)ka       * GAT_HD + n0 + mrow];
        bL.y = Wl[(size_t)(ka + 1) * GAT_HD + n0 + mrow];
        bR.x = Wr[(size_t)ka       * GAT_HD + n0 + mrow];
        bR.y = Wr[(size_t)(ka + 1) * GAT_HD + n0 + mrow];
        cl = __builtin_amdgcn_wmma_f32_16x16x4_f32(false, a, false, bL,
                                                   (short)0, cl, false, false);
        cr = __builtin_amdgcn_wmma_f32_16x16x4_f32(false, a, false, bR,
                                                   (short)0, cr, false, false);
    }

    const float addL = bl[n0 + mrow];
    const float addR = br[n0 + mrow];
    #pragma unroll
    for (int v = 0; v < 8; ++v) {
        const int m = m0 + v + 8 * grp;
        xl[(size_t)m * GAT_HD + n0 + mrow] = cl[v] + addL;
        xr[(size_t)m * GAT_HD + n0 + mrow] = cr[v] + addR;
    }
}

// ---------------------------------------------------------------------------
// Kernel 2: dense adjacency row -> compact neighbor list (<=32) + count.
// One wave per row; 16B non-temporal streaming loads (268 MB one-shot
// stream; keep it out of the way of L2-resident xl/xr); ballot+popc compact.
// ---------------------------------------------------------------------------
__global__ __launch_bounds__(256) void gat_nbr_extract(
    const float* __restrict__ adj,
    int* __restrict__ nbr, int* __restrict__ cnt)
{
    const int lane = threadIdx.x & 31;
    const int row  = blockIdx.x * 8 + (threadIdx.x >> 5);   // 0..16383
    const v4f* rp = (const v4f*)(adj + (size_t)row * GAT_N);

    int c = 0;
    for (int it = 0; it < GAT_N / 128; ++it) {              // 32 iterations
        const v4f v = __builtin_nontemporal_load(&rp[it * 32 + lane]);
        const int jbase = it * 128 + lane * 4;
        #pragma unroll
        for (int s = 0; s < 4; ++s) {
            const float val = v[s];
            const bool hit = (val > 0.0f);
            const unsigned mask = __builtin_amdgcn_ballot_w32(hit);
            if (hit) {
                const int pos = c + __popc(mask & ((1u << lane) - 1u));
                if (pos < GAT_K) nbr[(size_t)row * GAT_K + pos] = jbase + s;
            }
            c += __popc(mask);
        }
    }
    if (lane == 0) cnt[row] = (c > GAT_K) ? GAT_K : c;
}

// ---------------------------------------------------------------------------
// Kernel 3: GATv2 attention + segment softmax + aggregation + bias + LN.
// One block per node (128 threads = 4 waves, wave h = head h).
// Lane holds channels c0 = h*64+lane and c1 = c0+32.
// Edge list = cnt neighbors + self loop (always valid -> no -inf path).
// ---------------------------------------------------------------------------
__global__ __launch_bounds__(128) void gat_attn_ln(
    const float* __restrict__ xl, const float* __restrict__ xr,
    const int* __restrict__ nbr, const int* __restrict__ cnt,
    const float* __restrict__ att, const float* __restrict__ bias,
    const float* __restrict__ gamma, const float* __restrict__ beta,
    float* __restrict__ out)
{
    const int node  = blockIdx.x;              // b*N + i
    const int tid   = threadIdx.x;
    const int h     = tid >> 5;
    const int lane  = tid & 31;
    const int bbase = (node / GAT_N) * GAT_N;  // b*N
    const int c0    = h * GAT_D + lane;
    const int c1    = c0 + 32;

    __shared__ float e_s[GAT_H][40];
    __shared__ float red_s[2 * GAT_H];

    const float xr0 = xr[(size_t)node * GAT_HD + c0];
    const float xr1 = xr[(size_t)node * GAT_HD + c1];
    const float a0  = att[c0];                 // att flat [H*D] == channel idx
    const float a1  = att[c1];
    const int   cn  = cnt[node];
    const int   E   = cn + 1;                  // + self loop

    // ---- pass 1: attention logits e[h][k] -------------------------------
    for (int k = 0; k < E; ++k) {
        const int gj = (k < cn) ? (bbase + nbr[(size_t)node * GAT_K + k]) : node;
        float s0 = xl[(size_t)gj * GAT_HD + c0] + xr0;
        float s1 = xl[(size_t)gj * GAT_HD + c1] + xr1;
        s0 = (s0 > 0.0f) ? s0 : NEG_SLOPE * s0;
        s1 = (s1 > 0.0f) ? s1 : NEG_SLOPE * s1;
        float t = a0 * s0 + a1 * s1;
        #pragma unroll
        for (int off = 16; off > 0; off >>= 1) t += __shfl_xor(t, off, 32);
        if (lane == 0) e_s[h][k] = t;
    }
    __syncthreads();

    // ---- segment softmax over edges (per head / per wave) ---------------
    const float NEGBIG = -3.4e38f;
    const float eA = (lane      < E) ? e_s[h][lane]      : NEGBIG;
    const float eB = (lane + 32 < E) ? e_s[h][lane + 32] : NEGBIG;
    float m = fmaxf(eA, eB);
    #pragma unroll
    for (int off = 16; off > 0; off >>= 1) m = fmaxf(m, __shfl_xor(m, off, 32));
    const float pA = (lane      < E) ? __expf(eA - m) : 0.0f;
    const float pB = (lane + 32 < E) ? __expf(eB - m) : 0.0f;
    float s = pA + pB;
    #pragma unroll
    for (int off = 16; off > 0; off >>= 1) s += __shfl_xor(s, off, 32);
    const float inv = 1.0f / s;
    if (lane      < E) e_s[h][lane]      = pA * inv;
    if (lane + 32 < E) e_s[h][lane + 32] = pB * inv;
    __syncthreads();

    // ---- pass 2: weighted aggregation -----------------------------------
    float acc0 = 0.0f, acc1 = 0.0f;
    for (int k = 0; k < E; ++k) {
        const int gj = (k < cn) ? (bbase + nbr[(size_t)node * GAT_K + k]) : node;
        const float al = e_s[h][k];
        acc0 += al * xl[(size_t)gj * GAT_HD + c0];
        acc1 += al * xl[(size_t)gj * GAT_HD + c1];
    }
    const float o0 = acc0 + bias[c0];
    const float o1 = acc1 + bias[c1];

    // ---- LayerNorm over HD=256 channels (block reduction) ---------------
    float ps = o0 + o1;
    float pq = o0 * o0 + o1 * o1;
    #pragma unroll
    for (int off = 16; off > 0; off >>= 1) {
        ps += __shfl_xor(ps, off, 32);
        pq += __shfl_xor(pq, off, 32);
    }
    if (lane == 0) { red_s[h] = ps; red_s[GAT_H + h] = pq; }
    __syncthreads();
    const float S  = red_s[0] + red_s[1] + red_s[2] + red_s[3];
    const float Q  = red_s[4] + red_s[5] + red_s[6] + red_s[7];
    const float mu  = S * (1.0f / GAT_HD);
    const float var = Q * (1.0f / GAT_HD) - mu * mu;
    const float rst = rsqrtf(var + LN_EPS);
    out[(size_t)node * GAT_HD + c0] = (o0 - mu) * rst * gamma[c0] + beta[c0];
    out[(size_t)node * GAT_HD + c1] = (o1 - mu) * rst * gamma[c1] + beta[c1];
}

// ---------------------------------------------------------------------------
extern "C" void kernel_launch(void* const* d_in, const int* in_sizes, int n_in,
                              void* d_out, int out_size, void* d_ws, size_t ws_size,
                              hipStream_t stream) {
    const float* x     = (const float*)d_in[0];
    const float* adj   = (const float*)d_in[1];
    const float* Wl    = (const float*)d_in[2];
    const float* bl    = (const float*)d_in[3];
    const float* Wr    = (const float*)d_in[4];
    const float* br    = (const float*)d_in[5];
    const float* att   = (const float*)d_in[6];
    const float* bias  = (const float*)d_in[7];
    const float* gamma = (const float*)d_in[8];
    const float* beta  = (const float*)d_in[9];
    float* out = (float*)d_out;

    // workspace layout
    float* xl  = (float*)d_ws;                       // 16384*256 f32 = 16.78 MB
    float* xr  = xl + (size_t)GAT_NN * GAT_HD;       // 16.78 MB
    int*   nbr = (int*)(xr + (size_t)GAT_NN * GAT_HD);   // 16384*32 i32 = 2 MB
    int*   cnt = nbr + (size_t)GAT_NN * GAT_K;       // 64 KB

    // 1) fp32 WMMA projections (independent of kernel 2)
    gat_proj_wmma<<<dim3(GAT_NN / 16, GAT_HD / 16), 32, 0, stream>>>(
        x, Wl, bl, Wr, br, xl, xr);

    // 2) adjacency scan -> neighbor lists (one wave per row)
    gat_nbr_extract<<<GAT_NN / 8, 256, 0, stream>>>(adj, nbr, cnt);

    // 3) attention + softmax + aggregate + bias + layernorm
    gat_attn_ln<<<GAT_NN, 128, 0, stream>>>(
        xl, xr, nbr, cnt, att, bias, gamma, beta, out);
}